// RNN_12249246728246
// MI455X (gfx1250) — compile-verified
//
#include <hip/hip_runtime.h>

// MI455X / gfx1250, wave32. Fully fused LSTM (B=512, T=2048, I=3, H=16) + linear head.
// One wave owns 16 batch rows; recurrence runs entirely in registers using
// v_wmma_f32_16x16x4_f32; h is re-laid-out C->A each step via a 1KB LDS transpose.

typedef __attribute__((ext_vector_type(2))) float v2f;
typedef __attribute__((ext_vector_type(8))) float v8f;

#define T_LEN 2048
#define LDS_STRIDE 17   // pad 16->17 words: conflict-free transpose loads

__device__ __forceinline__ float fexp2(float x) { return __builtin_amdgcn_exp2f(x); }
__device__ __forceinline__ float frcp (float x) { return __builtin_amdgcn_rcpf(x);  }
// sigmoid(x) = 1/(1+2^(-x*log2 e))
__device__ __forceinline__ float sigm_(float x) { return frcp(1.0f + fexp2(x * -1.44269504088896340736f)); }
// tanh(x) = 1 - 2/(2^(2x*log2 e)+1)
__device__ __forceinline__ float tanh_(float x) { return 1.0f - 2.0f * frcp(1.0f + fexp2(x * 2.88539008177792681472f)); }

__global__ __launch_bounds__(32)
void lstm_fused_wmma(const float* __restrict__ x,      // [B,T,3]
                     const float* __restrict__ W_ih,   // [64,3]
                     const float* __restrict__ W_hh,   // [64,16]
                     const float* __restrict__ b_ih,   // [64]
                     const float* __restrict__ b_hh,   // [64]
                     const float* __restrict__ W_out,  // [3,16]
                     const float* __restrict__ b_out,  // [3]
                     float* __restrict__ out)          // [B,T,3]
{
    __shared__ float hlds[16 * LDS_STRIDE];

    const int lane = threadIdx.x & 31;
    const int lo   = lane & 15;     // N / batch-in-A / hidden-in-C position
    const int hi   = lane >> 4;     // half-wave selector
    const int b0   = blockIdx.x * 16;

    // ---- Combined gate B matrix, resident in VGPRs: Wc[K=20][N=64] ----
    // K 0..15 : W_hh^T   (h path)
    // K 16..18: W_ih^T   (x path)
    // K 19    : b_ih+b_hh (paired with constant-1 A element)
    // v_wmma_f32_16x16x4 B layout (mirror of A): VGPRq, lanes0-15 -> K=4k+q,
    // lanes16-31 -> K=4k+2+q, N = lane%16.
    v2f wB[4][5];
#pragma unroll
    for (int n = 0; n < 4; ++n)
#pragma unroll
        for (int k = 0; k < 5; ++k)
#pragma unroll
            for (int q = 0; q < 2; ++q) {
                const int K = 4 * k + 2 * hi + q;
                const int N = 16 * n + lo;
                float v;
                if      (K < 16) v = W_hh[N * 16 + K];
                else if (K < 19) v = W_ih[N * 3 + (K - 16)];
                else             v = b_ih[N] + b_hh[N];
                wB[n][k][q] = v;
            }

    // ---- Head B matrix: W_out^T zero-padded to 16x16 ----
    v2f hB[4];
#pragma unroll
    for (int k = 0; k < 4; ++k)
#pragma unroll
        for (int q = 0; q < 2; ++q) {
            const int K = 4 * k + 2 * hi + q;
            hB[k][q] = (lo < 3) ? W_out[lo * 16 + K] : 0.0f;
        }
    const float obias = (lo < 3) ? b_out[lo] : 0.0f;

    // Per-lane LDS transpose offsets (words).
    const int st_off = (8 * hi) * LDS_STRIDE + lo;  // + r*LDS_STRIDE  (batch-major store)
    const int ld_off = lo * LDS_STRIDE + 2 * hi;    // + 4k + q        (A-layout gather)

    // x stream for this lane: lanes0-15 supply (x0,x1), lanes16-31 supply (x2, 1.0)
    const float* xp = x + (b0 + lo) * (T_LEN * 3) + (hi ? 2 : 0);

    // Recurrent state (registers only)
    v8f cst;
    v2f ah[4];
#pragma unroll
    for (int r = 0; r < 8; ++r) cst[r] = 0.0f;
#pragma unroll
    for (int k = 0; k < 4; ++k) { ah[k][0] = 0.0f; ah[k][1] = 0.0f; }

    // Prime the x pipeline (t = 0)
    float xa = xp[0];
    float xb = 1.0f;
    if (!hi) xb = xp[1];

    const int out_base = (b0 + 8 * hi) * (T_LEN * 3) + lo;
    const bool do_out  = (lo < 3);

    for (int t = 0; t < T_LEN; ++t) {
        // ---- prefetch x for t+1 (hides global latency under this step) ----
        const int tn = (t + 1 < T_LEN) ? (t + 1) : t;
        float na = xp[tn * 3];
        float nb = 1.0f;
        if (!hi) nb = xp[tn * 3 + 1];

        v2f ax; ax[0] = xa; ax[1] = xb;

        // ---- gates = [h_{t-1} | x_t | 1] x Wc : 20 WMMAs ----
        v8f acc[4];
#pragma unroll
        for (int n = 0; n < 4; ++n) {
            v8f a = {};
#pragma unroll
            for (int k = 0; k < 4; ++k)
                a = __builtin_amdgcn_wmma_f32_16x16x4_f32(
                        false, ah[k], false, wB[n][k], (short)0, a, false, false);
            a = __builtin_amdgcn_wmma_f32_16x16x4_f32(
                        false, ax, false, wB[n][4], (short)0, a, false, false);
            acc[n] = a;
        }

        // ---- pointwise LSTM cell (lane = hidden unit, VGPR = batch row) ----
        v8f hval;
#pragma unroll
        for (int r = 0; r < 8; ++r) {
            const float iv = sigm_(acc[0][r]);
            const float fv = sigm_(acc[1][r]);
            const float gv = tanh_(acc[2][r]);
            const float ov = sigm_(acc[3][r]);
            const float cc = fv * cst[r] + iv * gv;
            cst[r] = cc;
            hval[r] = ov * tanh_(cc);
        }

        // ---- C-layout -> A-layout transpose through LDS (intra-wave) ----
#pragma unroll
        for (int r = 0; r < 8; ++r)
            hlds[st_off + r * LDS_STRIDE] = hval[r];
        __syncthreads();   // single-wave WG: lowers to dscnt wait + S_NOP
#pragma unroll
        for (int k = 0; k < 4; ++k) {
            ah[k][0] = hlds[ld_off + 4 * k + 0];
            ah[k][1] = hlds[ld_off + 4 * k + 1];
        }

        // ---- fused head: out_t = h_t x W_out^T + b_out : 4 WMMAs ----
        v8f oacc;
#pragma unroll
        for (int r = 0; r < 8; ++r) oacc[r] = obias;
#pragma unroll
        for (int k = 0; k < 4; ++k)
            oacc = __builtin_amdgcn_wmma_f32_16x16x4_f32(
                       false, ah[k], false, hB[k], (short)0, oacc, false, false);

        if (do_out) {
#pragma unroll
            for (int r = 0; r < 8; ++r)
                out[out_base + r * (T_LEN * 3) + t * 3] = oacc[r];
        }

        xa = na; xb = nb;
    }
}

extern "C" void kernel_launch(void* const* d_in, const int* in_sizes, int n_in,
                              void* d_out, int out_size, void* d_ws, size_t ws_size,
                              hipStream_t stream) {
    const float* x     = (const float*)d_in[0];
    const float* W_ih  = (const float*)d_in[1];
    const float* W_hh  = (const float*)d_in[2];
    const float* b_ih  = (const float*)d_in[3];
    const float* b_hh  = (const float*)d_in[4];
    const float* W_out = (const float*)d_in[5];
    const float* b_out = (const float*)d_in[6];
    float* out = (float*)d_out;

    const int B = in_sizes[0] / (T_LEN * 3);   // 512
    const int grid = B / 16;                   // 32 waves, one 16-batch chain each

    lstm_fused_wmma<<<grid, 32, 0, stream>>>(x, W_ih, W_hh, b_ih, b_hh,
                                             W_out, b_out, out);
}